// RetinaNet_58042188038696
// MI455X (gfx1250) — compile-verified
//
#include <hip/hip_runtime.h>
#include <hip/hip_bf16.h>
#include <stdint.h>

// RetinaNet post-processing for MI455X (gfx1250, wave32).
// Pipeline: decode -> per-class histogram top-K select -> async-DMA candidate
// gather (global_load_async_to_lds_b32) -> bitonic sort -> WMMA-assisted
// pairwise-IoU suppression matrix -> exact greedy NMS -> packed atomicMax
// merge -> finalize.
// v_wmma_f32_16x16x4_f32 computes the pairwise union denominator
// (area_i + area_j) as a rank-2 outer-product matmul.

#define NUM_CLASSES 80
#define TOP_K       300
#define CLS_THRES   0.05f
#define IOU_THR     0.5f
#define CAP         4096      // max candidates kept per class (superset of top-300)
#define NBINS       128
#define PAD         304       // TOP_K padded to 19*16
#define NTILE       19        // PAD/16
#define COLW        10        // ceil(PAD/32) bitmask words per column

typedef float v2f __attribute__((ext_vector_type(2)));
typedef float v8f __attribute__((ext_vector_type(8)));

__global__ void zero_kernel(unsigned* __restrict__ p, int n) {
    int i = blockIdx.x * blockDim.x + threadIdx.x;
    if (i < n) p[i] = 0u;
}

__global__ void decode_kernel(const float* __restrict__ anchors,
                              const float* __restrict__ regs,
                              const int* __restrict__ imgh,
                              const int* __restrict__ imgw,
                              float* __restrict__ boxes, int A) {
    int a = blockIdx.x * blockDim.x + threadIdx.x;
    if (a >= A) return;
    float a0 = anchors[4*a+0], a1 = anchors[4*a+1];
    float a2 = anchors[4*a+2], a3 = anchors[4*a+3];
    float w  = a2 - a0, h = a3 - a1;
    float cx = a0 + 0.5f * w, cy = a1 + 0.5f * h;
    float d0 = regs[4*a+0] * 0.1f, d1 = regs[4*a+1] * 0.1f;
    float d2 = regs[4*a+2] * 0.2f, d3 = regs[4*a+3] * 0.2f;
    float pcx = cx + d0 * w, pcy = cy + d1 * h;
    float pw  = expf(d2) * w, ph = expf(d3) * h;
    float W = (float)imgw[0], H = (float)imgh[0];
    boxes[4*a+0] = fmaxf(pcx - 0.5f * pw, 0.0f);
    boxes[4*a+1] = fmaxf(pcy - 0.5f * ph, 0.0f);
    boxes[4*a+2] = fminf(pcx + 0.5f * pw, W);
    boxes[4*a+3] = fminf(pcy + 0.5f * ph, H);
}

// Per-class histogram of above-threshold scores (coalesced flat streaming).
__global__ void hist_kernel(const float* __restrict__ cls, int N,
                            unsigned* __restrict__ ghist) {
    __shared__ unsigned lh[NUM_CLASSES * NBINS];
    for (int t = threadIdx.x; t < NUM_CLASSES * NBINS; t += blockDim.x) lh[t] = 0u;
    __syncthreads();
    int stride = gridDim.x * blockDim.x;
    for (int i = blockIdx.x * blockDim.x + threadIdx.x; i < N; i += stride) {
        __builtin_prefetch(cls + i + stride, 0, 1);   // global_prefetch_b8
        float s = cls[i];
        if (s > CLS_THRES) {
            int c = i % NUM_CLASSES;
            int b = (int)(s * (float)NBINS);
            b = b < 0 ? 0 : (b > NBINS - 1 ? NBINS - 1 : b);
            atomicAdd(&lh[c * NBINS + b], 1u);
        }
    }
    __syncthreads();
    for (int t = threadIdx.x; t < NUM_CLASSES * NBINS; t += blockDim.x)
        if (lh[t]) atomicAdd(&ghist[t], lh[t]);
}

// Find per-class cutoff bin: largest bin boundary v such that #(scores >= v) >= TOP_K.
__global__ void cutoff_kernel(const unsigned* __restrict__ ghist,
                              float* __restrict__ thr) {
    int c = threadIdx.x;
    if (c >= NUM_CLASSES) return;
    int acc = 0, b;
    for (b = NBINS - 1; b >= 0; --b) {
        acc += (int)ghist[c * NBINS + b];
        if (acc >= TOP_K) break;
    }
    thr[c] = (b < 0) ? 0.0f : ((float)b / (float)NBINS);
}

// Collect all candidates >= cutoff (superset of exact top-K).
__global__ void collect_kernel(const float* __restrict__ cls, int N,
                               const float* __restrict__ thr,
                               unsigned* __restrict__ cnt,
                               float* __restrict__ candS,
                               int* __restrict__ candI) {
    int stride = gridDim.x * blockDim.x;
    for (int i = blockIdx.x * blockDim.x + threadIdx.x; i < N; i += stride) {
        float s = cls[i];
        int c = i % NUM_CLASSES;
        if (s > CLS_THRES && s >= thr[c]) {
            unsigned p = atomicAdd(&cnt[c], 1u);
            if (p < CAP) {
                candS[c * CAP + p] = s;
                candI[c * CAP + p] = i / NUM_CLASSES;
            }
        }
    }
}

// One workgroup (8 wave32) per class: async gather, sort, suppression matrix
// (WMMA), greedy NMS.
__global__ void nms_kernel(const float* __restrict__ boxes,
                           const unsigned* __restrict__ cnt,
                           const float* __restrict__ candSg,
                           const int* __restrict__ candIg,
                           unsigned long long* __restrict__ packed) {
    __shared__ float sS[CAP];
    __shared__ int   idS[CAP];
    __shared__ float x1S[PAD], y1S[PAD], x2S[PAD], y2S[PAD], arS[PAD];
    __shared__ unsigned colbits[PAD * COLW];   // column-major suppression bits
    __shared__ int supS[PAD];
    __shared__ int keepS[TOP_K];

    int c = blockIdx.x;
    int tid = threadIdx.x;
    int m = (int)cnt[c]; if (m > CAP) m = CAP;

    // Candidate gather: async DMA global->LDS for the live prefix (ASYNCcnt
    // path, bypasses VGPRs); DS stores pad the tail.
    {
        const float* gs = candSg + (size_t)c * CAP;
        const int*   gi = candIg + (size_t)c * CAP;
        for (int t = tid; t < CAP; t += 256) {
            if (t < m) {
                unsigned ldsS = (unsigned)(uintptr_t)(&sS[t]);
                unsigned ldsI = (unsigned)(uintptr_t)(&idS[t]);
                unsigned off  = (unsigned)t * 4u;
                asm volatile("global_load_async_to_lds_b32 %0, %1, %2"
                             :: "v"(ldsS), "v"(off), "s"(gs) : "memory");
                asm volatile("global_load_async_to_lds_b32 %0, %1, %2"
                             :: "v"(ldsI), "v"(off), "s"(gi) : "memory");
            } else {
                sS[t]  = -1e30f;
                idS[t] = 0x7fffffff;
            }
        }
        asm volatile("s_wait_asynccnt 0x0" ::: "memory");
    }
    __syncthreads();

    // Bitonic sort: descending by score, ties by ascending anchor index.
    for (int k = 2; k <= CAP; k <<= 1) {
        for (int j = k >> 1; j > 0; j >>= 1) {
            for (int idx = tid; idx < CAP; idx += 256) {
                int ixj = idx ^ j;
                if (ixj > idx) {
                    bool dir = ((idx & k) == 0);  // true -> descending block
                    float s1 = sS[idx], s2 = sS[ixj];
                    int   i1 = idS[idx], i2 = idS[ixj];
                    bool worse = (s1 < s2) || (s1 == s2 && i1 > i2);
                    if (worse == dir) {
                        sS[idx] = s2; sS[ixj] = s1;
                        idS[idx] = i2; idS[ixj] = i1;
                    }
                }
            }
            __syncthreads();
        }
    }

    // Load top-K candidate boxes + areas into LDS (padding -> zero box).
    for (int t = tid; t < PAD; t += 256) {
        bool valid = (t < TOP_K) && (sS[t] > CLS_THRES);
        float b0 = 0.f, b1 = 0.f, b2 = 0.f, b3 = 0.f;
        if (valid) {
            int a = idS[t];
            b0 = boxes[4*a+0]; b1 = boxes[4*a+1];
            b2 = boxes[4*a+2]; b3 = boxes[4*a+3];
        }
        x1S[t] = b0; y1S[t] = b1; x2S[t] = b2; y2S[t] = b3;
        arS[t] = fmaxf(b2 - b0, 0.f) * fmaxf(b3 - b1, 0.f);
        supS[t] = 0;
    }
    for (int t = tid; t < PAD * COLW; t += 256) colbits[t] = 0u;
    __syncthreads();

    // Suppression matrix: union denominator (area_i + area_j) via
    // v_wmma_f32_16x16x4_f32 rank-2 outer-product; intersection via VALU.
    int wave = tid >> 5, lane = tid & 31;
    int half = lane >> 4, lm = lane & 15;
    float maskf = (half == 0) ? 1.0f : 0.0f;
    for (int t = wave; t < NTILE * NTILE; t += 8) {   // full-wave uniform loop: EXEC all ones
        int ti = t / NTILE, tj = t % NTILE;
        v2f av, bv;
        av[0] = arS[ti * 16 + lm] * maskf;  av[1] = maskf;          // A = [area_i, 1, 0, 0]
        bv[0] = maskf;  bv[1] = arS[tj * 16 + lm] * maskf;          // B = [[1..],[area_j..],0,0]
        v8f acc = {0.f, 0.f, 0.f, 0.f, 0.f, 0.f, 0.f, 0.f};
        acc = __builtin_amdgcn_wmma_f32_16x16x4_f32(
            false, av, false, bv, (short)0, acc, false, false);     // D[i][j] = area_i + area_j

        int j = tj * 16 + lm;
        int ibase = ti * 16 + half * 8;
        float xj1 = x1S[j], yj1 = y1S[j], xj2 = x2S[j], yj2 = y2S[j];
        unsigned field = 0u;
#pragma unroll
        for (int r = 0; r < 8; ++r) {
            int i = ibase + r;
            float iw = fminf(x2S[i], xj2) - fmaxf(x1S[i], xj1);
            float ih = fminf(y2S[i], yj2) - fmaxf(y1S[i], yj1);
            iw = fmaxf(iw, 0.f); ih = fmaxf(ih, 0.f);
            float inter = iw * ih;
            float denom = acc[r] - inter + 1e-8f;    // area_i + area_j - inter + eps
            if (inter > IOU_THR * denom)             // IoU > 0.5
                field |= (1u << ((ibase & 31) + r));
        }
        if (field) atomicOr(&colbits[j * COLW + (ibase >> 5)], field);
    }
    __syncthreads();

    // Exact sequential greedy NMS (parallel suppression per step).
    for (int i = 0; i < TOP_K; ++i) {
        if (tid == 0) keepS[i] = (!supS[i]) && (sS[i] > CLS_THRES);
        __syncthreads();
        if (keepS[i]) {
            unsigned w = (unsigned)(i >> 5), bit = (unsigned)(i & 31);
            for (int j = tid; j < PAD; j += 256)
                if (j > i && ((colbits[j * COLW + w] >> bit) & 1u)) supS[j] = 1;
        }
        __syncthreads();
    }

    // Merge kept detections: pack (score_bits<<32)|(79-c); max => highest score,
    // ties resolved to the lowest class (matches argmax-first semantics).
    for (int t = tid; t < TOP_K; t += 256) {
        if (keepS[t]) {
            int a = idS[t];
            unsigned long long pk =
                ((unsigned long long)__float_as_uint(sS[t]) << 32) |
                (unsigned long long)(unsigned)(NUM_CLASSES - 1 - c);
            atomicMax(&packed[a], pk);
        }
    }
}

__global__ void finalize_kernel(const unsigned long long* __restrict__ packed,
                                int A, float* __restrict__ out) {
    int a = blockIdx.x * blockDim.x + threadIdx.x;
    if (a >= A) return;
    float* scores = out;
    float* labels = out + A;
    float* boxes  = out + 2 * (size_t)A;
    unsigned long long p = packed[a];
    if (p != 0ull) {
        scores[a] = __uint_as_float((unsigned)(p >> 32));
        labels[a] = (float)(NUM_CLASSES - 1 - (int)(unsigned)(p & 0xffffffffull));
        // boxes already hold decoded clipped values from decode_kernel
    } else {
        scores[a] = 0.0f;
        labels[a] = -1.0f;
        boxes[4*a+0] = 0.f; boxes[4*a+1] = 0.f;
        boxes[4*a+2] = 0.f; boxes[4*a+3] = 0.f;
    }
}

extern "C" void kernel_launch(void* const* d_in, const int* in_sizes, int n_in,
                              void* d_out, int out_size, void* d_ws, size_t ws_size,
                              hipStream_t stream) {
    const float* cls     = (const float*)d_in[0];   // (1, A, 80) f32
    const float* regs    = (const float*)d_in[1];   // (1, A, 4)  f32
    const float* anchors = (const float*)d_in[2];   // (1, A, 4)  f32
    const int*   imgh    = (const int*)d_in[3];
    const int*   imgw    = (const int*)d_in[4];
    float* out = (float*)d_out;                     // [scores A][labels A][boxes A*4]

    int A = in_sizes[1] / 4;
    int N = in_sizes[0];                            // A * NUM_CLASSES

    // Workspace layout (u32 units): hist | cnt | thr | packed(u64) | candS | candI
    unsigned* ws32  = (unsigned*)d_ws;
    unsigned* hist  = ws32;                                          // 80*128
    unsigned* cnt   = ws32 + NUM_CLASSES * NBINS;                    // 80
    float*    thr   = (float*)(ws32 + NUM_CLASSES * NBINS + NUM_CLASSES);  // 80
    unsigned long long* packed =
        (unsigned long long*)(ws32 + NUM_CLASSES * NBINS + 2 * NUM_CLASSES); // A u64 (8B aligned)
    float* candS = (float*)(packed + A);                             // 80*CAP
    int*   candI = (int*)(candS + (size_t)NUM_CLASSES * CAP);        // 80*CAP
    float* boxes = out + 2 * (size_t)A;                              // decoded boxes live in d_out

    int zeroN = NUM_CLASSES * NBINS + 2 * NUM_CLASSES + 2 * A;       // hist+cnt+thr+packed
    zero_kernel<<<(zeroN + 255) / 256, 256, 0, stream>>>(ws32, zeroN);
    decode_kernel<<<(A + 255) / 256, 256, 0, stream>>>(anchors, regs, imgh, imgw, boxes, A);
    hist_kernel<<<512, 256, 0, stream>>>(cls, N, hist);
    cutoff_kernel<<<1, 128, 0, stream>>>(hist, thr);
    collect_kernel<<<512, 256, 0, stream>>>(cls, N, thr, cnt, candS, candI);
    nms_kernel<<<NUM_CLASSES, 256, 0, stream>>>(boxes, cnt, candS, candI, packed);
    finalize_kernel<<<(A + 255) / 256, 256, 0, stream>>>(packed, A, out);
}